// ArSpectralDiffusionTransformer_76785425318187
// MI455X (gfx1250) — compile-verified
//
#include <hip/hip_runtime.h>
#include <math.h>

#define Bz      4
#define Hh      32
#define Wd      17
#define Dd      512
#define NHEADS  8
#define NBLOCKS 12
#define HDs     64
#define SEQ     544          // Hh*Wd
#define NTOK    (Bz*SEQ)     // 2176

typedef __attribute__((ext_vector_type(2))) float v2f;
typedef __attribute__((ext_vector_type(8))) float v8f;

// ---------------------------------------------------------------------------
// U_IDS / IU_IDS permutation (port of _u_ids), computed on device each call.
// ---------------------------------------------------------------------------
__global__ void setup_ids_kernel(int* __restrict__ u, int* __restrict__ iu) {
    if (threadIdx.x != 0 || blockIdx.x != 0) return;
    int neg[16][17]; int pos[16][17];
    for (int r = 0; r < 16; ++r)
        for (int c = 0; c < 17; ++c) { neg[r][c] = 0; pos[r][c] = 0; }
    int start = 1;
    for (int i = 0; i < 17; ++i) {
        if (i < 16) {
            int n = (i + 1 < 17) ? (i + 1) : 17;
            for (int j = 0; j < n; ++j) neg[i][j] = start + j;
            start += n;
        }
        if (i + 1 < 17) {
            int n = (i + 1 < 16) ? (i + 1) : 16;
            for (int r = 0; r < n; ++r) neg[r][i + 1] = start + n - 1 - r;
            start += n;
        }
        if (i + 1 < 17) {
            int n = (i + 2 < 16) ? (i + 2) : 16;
            for (int r = 0; r < n; ++r) pos[r][i + 1] = start + r;
            start += n;
        }
        if (i + 1 < 16) {
            int n = (i + 1 < 17) ? (i + 1) : 17;
            for (int j = 0; j < n; ++j) pos[i + 1][j] = start + n - 1 - j;
            start += n;
        }
    }
    int idx = 0;
    for (int r = 0; r < 16; ++r) for (int c = 0; c < 17; ++c) u[idx++] = pos[r][c];
    for (int r = 15; r >= 0; --r) for (int c = 0; c < 17; ++c) u[idx++] = neg[r][c];
    for (int s = 0; s < SEQ; ++s) iu[u[s]] = s;
}

// ---------------------------------------------------------------------------
// Input projection (4 -> 512 complex) + positional embedding + U_IDS permute.
// ---------------------------------------------------------------------------
__global__ void input_proj_kernel(const float* __restrict__ xre, const float* __restrict__ xim,
                                  const float2* __restrict__ win, const float2* __restrict__ binb,
                                  const int* __restrict__ uids, float2* __restrict__ X) {
    int b = blockIdx.x / SEQ;
    int s = blockIdx.x % SEQ;
    int t = uids[s];
    int h = t / Wd, w = t % Wd;
    const float* xr = xre + ((size_t)(b * Hh * Wd) + t) * 4;
    const float* xi = xim + ((size_t)(b * Hh * Wd) + t) * 4;
    float ar[4], ai[4];
    for (int j = 0; j < 4; ++j) { ar[j] = xr[j]; ai[j] = xi[j]; }
    for (int d = threadIdx.x; d < Dd; d += 256) {
        float2 acc = binb[d];
        const float2* wrow = win + (size_t)d * 4;
#pragma unroll
        for (int j = 0; j < 4; ++j) {
            float2 wv = wrow[j];
            acc.x += ar[j] * wv.x - ai[j] * wv.y;
            acc.y += ar[j] * wv.y + ai[j] * wv.x;
        }
        float ang;
        if (d < Dd / 2) ang = (float)(h - Hh / 2) / __powf(10000.f, (float)d / (float)(Dd / 2));
        else            ang = (float)w            / __powf(10000.f, (float)(d - Dd / 2) / (float)(Dd / 2));
        acc.x += __cosf(ang);
        acc.y += __sinf(ang);
        X[(size_t)blockIdx.x * Dd + d] = acc;
    }
}

// ---------------------------------------------------------------------------
// Complex GEMM via V_WMMA_F32_16X16X4_F32.
//   C[m][n] = scale * sum_k A[m][k] * (CONJ ? conj(Bk) : Bk) (+bias[n]) (+res)
//   BTRANS=1: B element (k,n) = Bm[n*ldb + k]   (weights / key matrix)
//   BTRANS=0: B element (k,n) = Bm[k*ldb + n]   (value matrix)
// One wave computes one 16x32 output tile (2 N-tiles sharing the A fragment);
// 8 v8f accumulators; register double-buffering so fragment loads for k+4
// overlap the 8 WMMAs of step k. Batched over grid.y via (s1,s2,yDiv).
// ---------------------------------------------------------------------------
template <bool BTRANS, bool CONJ>
__global__ void cgemm_wmma(const float2* __restrict__ A, int lda, long long aS1, long long aS2,
                           const float2* __restrict__ Bm, int ldb, long long bS1, long long bS2,
                           const float2* __restrict__ bias, const float2* __restrict__ res,
                           float scale,
                           float2* __restrict__ C, int ldc, long long cS1, long long cS2,
                           int M, int N, int K, int yDiv) {
    int y = blockIdx.y;
    long long y1 = y / yDiv, y2 = y % yDiv;
    const float2* Ap = A + y1 * aS1 + y2 * aS2;
    const float2* Bp = Bm + y1 * bS1 + y2 * bS2;
    float2* Cp = C + y1 * cS1 + y2 * cS2;
    const float2* Rp = res ? (res + y1 * cS1 + y2 * cS2) : nullptr;

    int tilesN2 = N >> 5;                 // 16x32 tiles along N
    int tilesM  = M >> 4;
    int tile = blockIdx.x * (blockDim.x >> 5) + (threadIdx.x >> 5);
    if (tile >= tilesM * tilesN2) return; // wave-uniform: EXEC all-ones below
    int mt = tile / tilesN2, nt = tile % tilesN2;
    int m0 = mt << 4, n0 = nt << 5;

    int lane = threadIdx.x & 31;
    int half = lane >> 4;                 // 0: K={k,k+1}  1: K={k+2,k+3}
    int l16  = lane & 15;

    v8f aRR0 = {}, aRI0 = {}, aIR0 = {}, aII0 = {};
    v8f aRR1 = {}, aRI1 = {}, aIR1 = {}, aII1 = {};

    const float* Arow  = (const float*)(Ap + (size_t)(m0 + l16) * lda);
    const float* Brow0 = nullptr; const float* Brow1 = nullptr;
    if (BTRANS) {
        Brow0 = (const float*)(Bp + (size_t)(n0 + l16) * ldb);
        Brow1 = (const float*)(Bp + (size_t)(n0 + 16 + l16) * ldb);
    }
    int colB0 = n0 + l16, colB1 = n0 + 16 + l16;

    auto loadA = [&](int k, v2f& r, v2f& i) {
        float4 v = *(const float4*)(Arow + 2 * (size_t)(k + 2 * half));
        r[0] = v.x; r[1] = v.z; i[0] = v.y; i[1] = v.w;
    };
    auto loadBT = [&](const float* Brow, int k, v2f& r, v2f& i) {
        float4 v = *(const float4*)(Brow + 2 * (size_t)(k + 2 * half));
        r[0] = v.x; r[1] = v.z; i[0] = v.y; i[1] = v.w;
    };
    auto loadBN = [&](int col, int k, v2f& r, v2f& i) {
        int ka = k + 2 * half;
        float2 b0 = Bp[(size_t)ka * ldb + col];
        float2 b1 = Bp[(size_t)(ka + 1) * ldb + col];
        r[0] = b0.x; r[1] = b1.x; i[0] = b0.y; i[1] = b1.y;
    };

    v2f ar, ai, br0, bi0, br1, bi1;
    loadA(0, ar, ai);
    if (BTRANS) { loadBT(Brow0, 0, br0, bi0); loadBT(Brow1, 0, br1, bi1); }
    else        { loadBN(colB0, 0, br0, bi0); loadBN(colB1, 0, br1, bi1); }

    for (int k = 0; k < K - 4; k += 4) {
        v2f arN, aiN, br0N, bi0N, br1N, bi1N;
        loadA(k + 4, arN, aiN);
        if (BTRANS) { loadBT(Brow0, k + 4, br0N, bi0N); loadBT(Brow1, k + 4, br1N, bi1N); }
        else        { loadBN(colB0, k + 4, br0N, bi0N); loadBN(colB1, k + 4, br1N, bi1N); }

        aRR0 = __builtin_amdgcn_wmma_f32_16x16x4_f32(false, ar, false, br0, (short)0, aRR0, false, false);
        aRI0 = __builtin_amdgcn_wmma_f32_16x16x4_f32(false, ar, false, bi0, (short)0, aRI0, false, false);
        aIR0 = __builtin_amdgcn_wmma_f32_16x16x4_f32(false, ai, false, br0, (short)0, aIR0, false, false);
        aII0 = __builtin_amdgcn_wmma_f32_16x16x4_f32(false, ai, false, bi0, (short)0, aII0, false, false);
        aRR1 = __builtin_amdgcn_wmma_f32_16x16x4_f32(false, ar, false, br1, (short)0, aRR1, false, false);
        aRI1 = __builtin_amdgcn_wmma_f32_16x16x4_f32(false, ar, false, bi1, (short)0, aRI1, false, false);
        aIR1 = __builtin_amdgcn_wmma_f32_16x16x4_f32(false, ai, false, br1, (short)0, aIR1, false, false);
        aII1 = __builtin_amdgcn_wmma_f32_16x16x4_f32(false, ai, false, bi1, (short)0, aII1, false, false);

        ar = arN; ai = aiN; br0 = br0N; bi0 = bi0N; br1 = br1N; bi1 = bi1N;
    }
    // last K step
    aRR0 = __builtin_amdgcn_wmma_f32_16x16x4_f32(false, ar, false, br0, (short)0, aRR0, false, false);
    aRI0 = __builtin_amdgcn_wmma_f32_16x16x4_f32(false, ar, false, bi0, (short)0, aRI0, false, false);
    aIR0 = __builtin_amdgcn_wmma_f32_16x16x4_f32(false, ai, false, br0, (short)0, aIR0, false, false);
    aII0 = __builtin_amdgcn_wmma_f32_16x16x4_f32(false, ai, false, bi0, (short)0, aII0, false, false);
    aRR1 = __builtin_amdgcn_wmma_f32_16x16x4_f32(false, ar, false, br1, (short)0, aRR1, false, false);
    aRI1 = __builtin_amdgcn_wmma_f32_16x16x4_f32(false, ar, false, bi1, (short)0, aRI1, false, false);
    aIR1 = __builtin_amdgcn_wmma_f32_16x16x4_f32(false, ai, false, br1, (short)0, aIR1, false, false);
    aII1 = __builtin_amdgcn_wmma_f32_16x16x4_f32(false, ai, false, bi1, (short)0, aII1, false, false);

#pragma unroll
    for (int half_n = 0; half_n < 2; ++half_n) {
        v8f cr, ci;
        if (half_n == 0) {
            if (CONJ) { cr = aRR0 + aII0; ci = aIR0 - aRI0; }
            else      { cr = aRR0 - aII0; ci = aRI0 + aIR0; }
        } else {
            if (CONJ) { cr = aRR1 + aII1; ci = aIR1 - aRI1; }
            else      { cr = aRR1 - aII1; ci = aRI1 + aIR1; }
        }
        cr *= scale; ci *= scale;
        int col = n0 + 16 * half_n + l16;
        float2 bb = make_float2(0.f, 0.f);
        if (bias) bb = bias[col];
#pragma unroll
        for (int v = 0; v < 8; ++v) {
            int row = m0 + v + 8 * half;  // C/D layout: VGPR v -> M=v (+8 upper half)
            float ox = cr[v] + bb.x, oy = ci[v] + bb.y;
            if (Rp) { float2 rv = Rp[(size_t)row * ldc + col]; ox += rv.x; oy += rv.y; }
            Cp[(size_t)row * ldc + col] = make_float2(ox, oy);
        }
    }
}

// ---------------------------------------------------------------------------
// Complex softmax over one score row: w = exp(s - max(Re s)); w /= sum(w).
// ---------------------------------------------------------------------------
__global__ void attn_softmax_kernel(float2* __restrict__ S, int len) {
    float2* row = S + (size_t)blockIdx.x * len;
    int tid = threadIdx.x;
    __shared__ float r1[256], r2[256];
    float mx = -3.402823466e38f;
    for (int j = tid; j < len; j += 256) mx = fmaxf(mx, row[j].x);
    r1[tid] = mx; __syncthreads();
    for (int s = 128; s > 0; s >>= 1) { if (tid < s) r1[tid] = fmaxf(r1[tid], r1[tid + s]); __syncthreads(); }
    mx = r1[0]; __syncthreads();
    float sr = 0.f, si = 0.f;
    for (int j = tid; j < len; j += 256) {
        float2 v = row[j];
        float e = __expf(v.x - mx);
        float er = e * __cosf(v.y), ei = e * __sinf(v.y);
        row[j] = make_float2(er, ei);
        sr += er; si += ei;
    }
    r1[tid] = sr; r2[tid] = si; __syncthreads();
    for (int s = 128; s > 0; s >>= 1) { if (tid < s) { r1[tid] += r1[tid + s]; r2[tid] += r2[tid + s]; } __syncthreads(); }
    sr = r1[0]; si = r2[0];
    float inv = 1.f / (sr * sr + si * si);
    for (int j = tid; j < len; j += 256) {
        float2 v = row[j];
        row[j] = make_float2((v.x * sr + v.y * si) * inv, (v.y * sr - v.x * si) * inv);
    }
}

// ---------------------------------------------------------------------------
// Whitening complex layernorm (_sln), one block per token, in place.
// ---------------------------------------------------------------------------
__global__ void sln_kernel(float2* __restrict__ Xp, const float* __restrict__ gamma,
                           const float2* __restrict__ beta) {
    int tid = threadIdx.x;
    float2* row = Xp + (size_t)blockIdx.x * Dd;
    float2 v0 = row[tid], v1 = row[tid + 256];
    __shared__ float r1[256], r2[256], r3[256];
    r1[tid] = v0.x + v1.x; r2[tid] = v0.y + v1.y; __syncthreads();
    for (int s = 128; s > 0; s >>= 1) { if (tid < s) { r1[tid] += r1[tid + s]; r2[tid] += r2[tid + s]; } __syncthreads(); }
    float mr = r1[0] * (1.f / Dd), mi = r2[0] * (1.f / Dd);
    __syncthreads();
    v0.x -= mr; v0.y -= mi; v1.x -= mr; v1.y -= mi;
    r1[tid] = v0.x * v0.x + v1.x * v1.x;
    r2[tid] = v0.y * v0.y + v1.y * v1.y;
    r3[tid] = v0.x * v0.y + v1.x * v1.y;
    __syncthreads();
    for (int s = 128; s > 0; s >>= 1) {
        if (tid < s) { r1[tid] += r1[tid + s]; r2[tid] += r2[tid + s]; r3[tid] += r3[tid + s]; }
        __syncthreads();
    }
    const float invdm1 = 1.f / (float)(Dd - 1);
    float a  = r1[0] * invdm1 + 1e-5f;
    float c  = r2[0] * invdm1 + 1e-5f;
    float bo = r3[0] * invdm1;
    float s_ = sqrtf(fmaxf(a * c - bo * bo, 0.f));
    float t_ = sqrtf(a + c + 2.f * s_);
    float inv = 1.f / (s_ * t_);
    float i00 = (c + s_) * inv, i01 = -bo * inv, i11 = (a + s_) * inv;
    {
        float yr = i00 * v0.x + i01 * v0.y, yi = i01 * v0.x + i11 * v0.y;
        const float* g = gamma + 3 * tid; float2 bt = beta[tid];
        row[tid] = make_float2(g[0] * yr + g[1] * yi + bt.x, g[1] * yr + g[2] * yi + bt.y);
    }
    {
        int d = tid + 256;
        float yr = i00 * v1.x + i01 * v1.y, yi = i01 * v1.x + i11 * v1.y;
        const float* g = gamma + 3 * d; float2 bt = beta[d];
        row[d] = make_float2(g[0] * yr + g[1] * yi + bt.x, g[1] * yr + g[2] * yi + bt.y);
    }
}

// ---------------------------------------------------------------------------
// Component-wise complex softplus (beta=8 hybrid), in place.
// ---------------------------------------------------------------------------
__device__ __forceinline__ float softplus8(float x) {
    float bx = 8.f * x;
    if (bx > 1.f) return x;
    float m = bx > 0.f ? bx : 0.f;                 // logaddexp(0,bx) stable form
    return (m + log1pf(__expf(-fabsf(bx)))) * 0.125f;
}
__global__ void csoftplus_kernel(float2* __restrict__ p, int n) {
    int i = blockIdx.x * blockDim.x + threadIdx.x;
    if (i < n) { float2 v = p[i]; p[i] = make_float2(softplus8(v.x), softplus8(v.y)); }
}

// ---------------------------------------------------------------------------
// Inverse permute + final 512 -> 4 complex projection.
// ---------------------------------------------------------------------------
__global__ void out_proj_kernel(const float2* __restrict__ X, const int* __restrict__ iuids,
                                const float2* __restrict__ wout, const float2* __restrict__ boutb,
                                float2* __restrict__ out) {
    int b = blockIdx.x / SEQ;
    int t = blockIdx.x % SEQ;
    int src = iuids[t];
    const float2* xr = X + ((size_t)b * SEQ + src) * Dd;
    int tid = threadIdx.x;
    float accr[4] = {0, 0, 0, 0}, acci[4] = {0, 0, 0, 0};
    for (int d = tid; d < Dd; d += 256) {
        float2 xv = xr[d];
#pragma unroll
        for (int e = 0; e < 4; ++e) {
            float2 wv = wout[(size_t)e * Dd + d];
            accr[e] += xv.x * wv.x - xv.y * wv.y;
            acci[e] += xv.x * wv.y + xv.y * wv.x;
        }
    }
    __shared__ float r1[256], r2[256];
    for (int e = 0; e < 4; ++e) {
        r1[tid] = accr[e]; r2[tid] = acci[e]; __syncthreads();
        for (int s = 128; s > 0; s >>= 1) { if (tid < s) { r1[tid] += r1[tid + s]; r2[tid] += r2[tid + s]; } __syncthreads(); }
        if (tid == 0) {
            float2 bb = boutb[e];
            out[(size_t)blockIdx.x * 4 + e] = make_float2(r1[0] + bb.x, r2[0] + bb.y);
        }
        __syncthreads();
    }
}

// ---------------------------------------------------------------------------
// Host-side launcher helpers
// ---------------------------------------------------------------------------
static void cg(hipStream_t st, int btrans, int conj,
               const float2* A, int lda, long long aS1, long long aS2,
               const float2* Bm, int ldb, long long bS1, long long bS2,
               const float2* bias, const float2* res, float scale,
               float2* C, int ldc, long long cS1, long long cS2,
               int M, int N, int K, int batches, int yDiv) {
    int tiles = (M >> 4) * (N >> 5);       // 16x32 tiles per wave
    dim3 grid((tiles + 7) / 8, batches);
    if (btrans && conj)
        cgemm_wmma<true, true><<<grid, 256, 0, st>>>(A, lda, aS1, aS2, Bm, ldb, bS1, bS2,
                                                     bias, res, scale, C, ldc, cS1, cS2, M, N, K, yDiv);
    else if (btrans)
        cgemm_wmma<true, false><<<grid, 256, 0, st>>>(A, lda, aS1, aS2, Bm, ldb, bS1, bS2,
                                                      bias, res, scale, C, ldc, cS1, cS2, M, N, K, yDiv);
    else
        cgemm_wmma<false, false><<<grid, 256, 0, st>>>(A, lda, aS1, aS2, Bm, ldb, bS1, bS2,
                                                       bias, res, scale, C, ldc, cS1, cS2, M, N, K, yDiv);
}

extern "C" void kernel_launch(void* const* d_in, const int* in_sizes, int n_in,
                              void* d_out, int out_size, void* d_ws, size_t ws_size,
                              hipStream_t stream) {
    const float*  x_re = (const float*)d_in[0];
    const float*  x_im = (const float*)d_in[1];
    const float2* win  = (const float2*)d_in[2];
    const float2* bin  = (const float2*)d_in[3];
    const float2* wq   = (const float2*)d_in[4];
    const float2* bq   = (const float2*)d_in[5];
    const float2* wk   = (const float2*)d_in[6];
    const float2* bk   = (const float2*)d_in[7];
    const float2* wv   = (const float2*)d_in[8];
    const float2* bv   = (const float2*)d_in[9];
    const float2* wo   = (const float2*)d_in[10];
    const float2* bo   = (const float2*)d_in[11];
    const float*  g1   = (const float*)d_in[12];
    const float2* be1  = (const float2*)d_in[13];
    const float2* wf1  = (const float2*)d_in[14];
    const float2* bf1  = (const float2*)d_in[15];
    const float2* wf2  = (const float2*)d_in[16];
    const float2* bf2  = (const float2*)d_in[17];
    const float*  g2   = (const float*)d_in[18];
    const float2* be2  = (const float2*)d_in[19];
    const float2* wout = (const float2*)d_in[20];
    const float2* bout = (const float2*)d_in[21];
    float2* out = (float2*)d_out;

    // workspace layout (interleaved complex planes)
    size_t actN = (size_t)NTOK * Dd;                   // 2176*512 complex
    float2* X  = (float2*)d_ws;
    float2* Q  = X  + actN;
    float2* Kp = Q  + actN;
    float2* V  = Kp + actN;
    float2* AO = V  + actN;
    float2* Hf = AO + actN;
    float2* S  = Hf + actN;
    size_t sN = (size_t)Bz * NHEADS * SEQ * SEQ;       // scores
    int* uids  = (int*)(S + sN);
    int* iuids = uids + SEQ;

    setup_ids_kernel<<<1, 32, 0, stream>>>(uids, iuids);
    input_proj_kernel<<<NTOK, 256, 0, stream>>>(x_re, x_im, win, bin, uids, X);

    const long long sQD = (long long)SEQ * Dd;   // per-batch stride of act planes
    const long long sSS = (long long)SEQ * SEQ;  // per-head stride of score plane

    for (int i = 0; i < NBLOCKS; ++i) {
        const float2* wqi = wq + (size_t)i * Dd * Dd;  const float2* bqi = bq + (size_t)i * Dd;
        const float2* wki = wk + (size_t)i * Dd * Dd;  const float2* bki = bk + (size_t)i * Dd;
        const float2* wvi = wv + (size_t)i * Dd * Dd;  const float2* bvi = bv + (size_t)i * Dd;
        const float2* woi = wo + (size_t)i * Dd * Dd;  const float2* boi = bo + (size_t)i * Dd;
        const float2* wf1i = wf1 + (size_t)i * Dd * Dd; const float2* bf1i = bf1 + (size_t)i * Dd;
        const float2* wf2i = wf2 + (size_t)i * Dd * Dd; const float2* bf2i = bf2 + (size_t)i * Dd;
        const float*  g1i = g1 + (size_t)i * Dd * 3;    const float2* be1i = be1 + (size_t)i * Dd;
        const float*  g2i = g2 + (size_t)i * Dd * 3;    const float2* be2i = be2 + (size_t)i * Dd;

        // Q/K/V projections: (2176x512) x (512x512)^T, +bias
        cg(stream, 1, 0, X, Dd, 0, 0, wqi, Dd, 0, 0, bqi, nullptr, 1.f, Q,  Dd, 0, 0, NTOK, Dd, Dd, 1, 1);
        cg(stream, 1, 0, X, Dd, 0, 0, wki, Dd, 0, 0, bki, nullptr, 1.f, Kp, Dd, 0, 0, NTOK, Dd, Dd, 1, 1);
        cg(stream, 1, 0, X, Dd, 0, 0, wvi, Dd, 0, 0, bvi, nullptr, 1.f, V,  Dd, 0, 0, NTOK, Dd, Dd, 1, 1);

        // scores[b,h] = (Q_bh 544x64) * conj(K_bh)^T / 8
        cg(stream, 1, 1,
           Q,  Dd, sQD, HDs,
           Kp, Dd, sQD, HDs,
           nullptr, nullptr, 0.125f,
           S, SEQ, (long long)NHEADS * sSS, sSS,
           SEQ, SEQ, HDs, Bz * NHEADS, NHEADS);

        attn_softmax_kernel<<<Bz * NHEADS * SEQ, 256, 0, stream>>>(S, SEQ);

        // AO[b,h] = W_attn (544x544) * V_bh (544x64)
        cg(stream, 0, 0,
           S, SEQ, (long long)NHEADS * sSS, sSS,
           V, Dd, sQD, HDs,
           nullptr, nullptr, 1.f,
           AO, Dd, sQD, HDs,
           SEQ, HDs, SEQ, Bz * NHEADS, NHEADS);

        // output projection + residual -> X (in place)
        cg(stream, 1, 0, AO, Dd, 0, 0, woi, Dd, 0, 0, boi, X, 1.f, X, Dd, 0, 0, NTOK, Dd, Dd, 1, 1);
        sln_kernel<<<NTOK, 256, 0, stream>>>(X, g1i, be1i);

        // FFN: Hf = softplus(X*wf1 + bf1); X = Hf*wf2 + bf2 + X; sln
        cg(stream, 1, 0, X, Dd, 0, 0, wf1i, Dd, 0, 0, bf1i, nullptr, 1.f, Hf, Dd, 0, 0, NTOK, Dd, Dd, 1, 1);
        csoftplus_kernel<<<(int)((actN + 255) / 256), 256, 0, stream>>>(Hf, (int)actN);
        cg(stream, 1, 0, Hf, Dd, 0, 0, wf2i, Dd, 0, 0, bf2i, X, 1.f, X, Dd, 0, 0, NTOK, Dd, Dd, 1, 1);
        sln_kernel<<<NTOK, 256, 0, stream>>>(X, g2i, be2i);
    }

    out_proj_kernel<<<NTOK, 256, 0, stream>>>(X, iuids, wout, bout, out);

    (void)in_sizes; (void)n_in; (void)out_size; (void)ws_size;
}